// SoftGLORIAGlobalContrastiveLoss_24910810317416
// MI455X (gfx1250) — compile-verified
//
#include <hip/hip_runtime.h>
#include <hip/hip_bf16.h>
#include <math.h>

// CDNA5 (gfx1250) fused soft-GLORIA global contrastive loss.
// - scores via V_WMMA_F32_16X16X4_F32 (full f32 matrix pipe)
// - A-block staged into LDS via GLOBAL_LOAD_ASYNC_TO_LDS_B128 (ASYNCcnt path)
// - scores bounded (|cos|*10 <= 10) -> fixed-shift logsumexp, per-lane accum,
//   no per-strip cross-lane traffic.

typedef float v2f __attribute__((ext_vector_type(2)));
typedef float v8f __attribute__((ext_vector_type(8)));

#define EPS_   1e-8f
#define TEMP3_ 10.0f
#define SHIFT_ 10.0f

// Butterfly-sum within each 16-lane half, then return the value so that
// lane r picks up half0's total and lane r+8 picks up half1's total
// (all other lanes get 0). Summing over r=0..7 gives each lane<16 exactly
// the total for row (i0 + lane).
__device__ __forceinline__ float reduce16_pick(float v, int r, int lane) {
#pragma unroll
  for (int off = 8; off >= 1; off >>= 1) v += __shfl_xor(v, off);
  const float v1 = __shfl(v, 16);
  return (lane == r) ? v : ((lane == r + 8) ? v1 : 0.f);
}

// ---------------- row L2 norms: one wave per row ----------------
__global__ __launch_bounds__(256) void row_norm_kernel(const float* __restrict__ X,
                                                       float* __restrict__ nrm, int D) {
  const int lane = threadIdx.x & 31;
  const int wave = threadIdx.x >> 5;
  const int row  = blockIdx.x * 8 + wave;
  const float* p = X + (size_t)row * D;
  float s = 0.f;
  for (int k = lane * 4; k < D; k += 128) {
    float4 x = *(const float4*)(p + k);
    s += x.x * x.x + x.y * x.y + x.z * x.z + x.w * x.w;
  }
#pragma unroll
  for (int off = 16; off >= 1; off >>= 1) s += __shfl_xor(s, off);
  if (lane == 0) nrm[row] = sqrtf(s);
}

// ---------------- fused GEMM + masked row reduction ----------------
// PHASE 1: sum of exp(s-SHIFT) over negatives (+ optional pos/neg counts)
// PHASE 2: per-positive loss terms using precomputed lse / counts
template <int PHASE, bool COUNTS>
__global__ __launch_bounds__(128, 1) void fused_kernel(
    const float* __restrict__ A, const float* __restrict__ Bm,
    const float* __restrict__ normA, const float* __restrict__ normB,
    const float* __restrict__ idx, const float* __restrict__ probs,
    float* __restrict__ lse_out, const float* __restrict__ lse_in,
    float* __restrict__ nposArr, float* __restrict__ nnegArr,
    float* __restrict__ rowloss, int B, int D) {
  extern __shared__ float aLds[];  // 64 x D floats (128 KB for D=512; 320 KB LDS/WGP)
  const int tid  = threadIdx.x;
  const int lane = tid & 31;
  const int w    = tid >> 5;    // wave 0..3
  const int hl   = lane >> 4;   // half-wave 0/1
  const int ln   = lane & 15;
  const int rowBase = blockIdx.x * 64;

  // Stage the block's 64 A-rows into LDS once, via the async Global->LDS path
  // (GLOBAL_LOAD_ASYNC_TO_LDS_B128, GV mode: vdst = LDS byte addr, v[addr] = 64b
  //  global addr, saddr = off). Tracked by ASYNCcnt; bypasses the VGPR file.
  {
    const float4* src = (const float4*)(A + (size_t)rowBase * D);
    const int n4 = (64 * D) >> 2;
    for (int e = tid; e < n4; e += blockDim.x) {
      // low 32 bits of a generic LDS address == LDS byte offset (aperture rule)
      unsigned ldsAddr = (unsigned)(uintptr_t)(&aLds[(size_t)e * 4]);
      unsigned long long gAddr = (unsigned long long)(uintptr_t)(src + e);
      asm volatile("global_load_async_to_lds_b128 %0, %1, off"
                   :
                   : "v"(ldsAddr), "v"(gAddr)
                   : "memory");
    }
    asm volatile("s_wait_asynccnt 0x0" ::: "memory");
  }
  __syncthreads();

  const float th1 = probs[0], th2 = probs[1];
  const int i0 = rowBase + w * 16;

  float na[8], le8[8], idn8[8];
  const float* idxRow[8];
#pragma unroll
  for (int r = 0; r < 8; ++r) {
    const int row = i0 + r + 8 * hl;   // D-matrix layout: VGPR r holds M=r / M=r+8
    na[r] = normA[row];
    idxRow[r] = idx + (size_t)row * B;
    if constexpr (PHASE == 2) {
      le8[r]  = lse_in[row];
      idn8[r] = 1.f / (1.f + nnegArr[row]);
    } else {
      le8[r] = 0.f; idn8[r] = 0.f;
    }
  }

  // per-lane partial accumulators for the 8 rows this lane touches
  float sum8[8]  = {0.f, 0.f, 0.f, 0.f, 0.f, 0.f, 0.f, 0.f};
  float cPos8[8] = {0.f, 0.f, 0.f, 0.f, 0.f, 0.f, 0.f, 0.f};
  float cNeg8[8] = {0.f, 0.f, 0.f, 0.f, 0.f, 0.f, 0.f, 0.f};

  // A fragment base: lane ln -> M=ln, K pair selected by half (K = 2*hl, 2*hl+1)
  const float* aBase = aLds + (size_t)(w * 16 + ln) * D + 2 * hl;

  for (int j0 = 0; j0 < B; j0 += 64) {
    // prefetch the idx tile consumed by this strip's epilogue (global_prefetch_b8)
#pragma unroll
    for (int r = 0; r < 8; ++r) {
      __builtin_prefetch(idxRow[r] + j0 + ln, 0, 1);
      __builtin_prefetch(idxRow[r] + j0 + 32 + ln, 0, 1);
    }

    v8f acc[4] = {};
    float nb[4];
    const float* bptr[4];
#pragma unroll
    for (int nt = 0; nt < 4; ++nt) {
      const int j = j0 + nt * 16 + ln;
      nb[nt] = normB[j];
      bptr[nt] = Bm + (size_t)j * D + 2 * hl;  // B frag: lane ln -> N=ln, K pair by half
    }

#pragma unroll 2
    for (int kk = 0; kk < D; kk += 4) {
      v2f a = *(const v2f*)(aBase + kk);
#pragma unroll
      for (int nt = 0; nt < 4; ++nt) {
        v2f b = *(const v2f*)(bptr[nt] + kk);
        acc[nt] = __builtin_amdgcn_wmma_f32_16x16x4_f32(
            false, a, false, b, (short)0, acc[nt], false, false);
      }
    }

    // epilogue: scale to cosine*10, apply idx masks, per-lane accumulation only
#pragma unroll
    for (int nt = 0; nt < 4; ++nt) {
      const int j = j0 + nt * 16 + ln;
#pragma unroll
      for (int r = 0; r < 8; ++r) {
        const float sc = acc[nt][r] * (TEMP3_ / fmaxf(na[r] * nb[nt], EPS_));
        const float v  = idxRow[r][j];
        if constexpr (PHASE == 1) {
          if (v <= th2) {
            sum8[r] += __expf(sc - SHIFT_);   // bounded: sc in [-10,10]
            if constexpr (COUNTS) cNeg8[r] += 1.f;
          }
          if constexpr (COUNTS) {
            if (v > th1) cPos8[r] += 1.f;
          }
        } else {
          // l = log1p(exp(lse - s)) / (1 + n_neg)   for positives
          if (v > th1) sum8[r] += log1pf(__expf(le8[r] - sc)) * idn8[r];
        }
      }
    }
  }

  // single cross-lane reduction at the end (8 butterflies per quantity)
  float rowMain = 0.f, rowP = 0.f, rowN = 0.f;
#pragma unroll
  for (int r = 0; r < 8; ++r) {
    rowMain += reduce16_pick(sum8[r], r, lane);
    if constexpr (PHASE == 1 && COUNTS) {
      rowP += reduce16_pick(cPos8[r], r, lane);
      rowN += reduce16_pick(cNeg8[r], r, lane);
    }
  }

  if (lane < 16) {
    const int row = i0 + lane;
    if constexpr (PHASE == 1) {
      lse_out[row] = __logf(rowMain) + SHIFT_;  // empty neg set -> -inf (matches jax)
      if constexpr (COUNTS) { nposArr[row] = rowP; nnegArr[row] = rowN; }
    } else {
      rowloss[row] = rowMain / nposArr[row];
    }
  }
}

// ---------------- final scalar reduction (deterministic, no atomics) ----------------
__global__ __launch_bounds__(256) void finalize_kernel(const float* __restrict__ rl0,
                                                       const float* __restrict__ rl1,
                                                       float* __restrict__ out, int B) {
  __shared__ float red[8];
  float s = 0.f;
  for (int i = threadIdx.x; i < B; i += 256) s += rl0[i] + rl1[i];
#pragma unroll
  for (int off = 16; off >= 1; off >>= 1) s += __shfl_xor(s, off);
  const int lane = threadIdx.x & 31, wv = threadIdx.x >> 5;
  if (lane == 0) red[wv] = s;
  __syncthreads();
  if (threadIdx.x == 0) {
    float t = 0.f;
    for (int i = 0; i < 8; ++i) t += red[i];
    out[0] = t / (float)B;
  }
}

extern "C" void kernel_launch(void* const* d_in, const int* in_sizes, int n_in,
                              void* d_out, int out_size, void* d_ws, size_t ws_size,
                              hipStream_t stream) {
  (void)n_in; (void)out_size; (void)ws_size;
  const float* cnn   = (const float*)d_in[0];
  const float* rnn   = (const float*)d_in[1];
  const float* idx   = (const float*)d_in[2];
  const float* probs = (const float*)d_in[3];

  const int B = (int)llround(sqrt((double)in_sizes[2]));  // idx is B*B
  const int D = in_sizes[0] / B;                          // cnn is B*D

  // workspace layout: 8 row-arrays of B floats (256 KB for B=8192)
  float* ws   = (float*)d_ws;
  float* cn   = ws + 0 * (size_t)B;
  float* rn   = ws + 1 * (size_t)B;
  float* lse0 = ws + 2 * (size_t)B;
  float* lse1 = ws + 3 * (size_t)B;
  float* np   = ws + 4 * (size_t)B;
  float* nn   = ws + 5 * (size_t)B;
  float* rl0  = ws + 6 * (size_t)B;
  float* rl1  = ws + 7 * (size_t)B;

  const size_t lds = (size_t)64 * D * sizeof(float);  // 128 KB for D=512
  (void)hipFuncSetAttribute((const void*)fused_kernel<1, true>,
                            hipFuncAttributeMaxDynamicSharedMemorySize, (int)lds);
  (void)hipFuncSetAttribute((const void*)fused_kernel<1, false>,
                            hipFuncAttributeMaxDynamicSharedMemorySize, (int)lds);
  (void)hipFuncSetAttribute((const void*)fused_kernel<2, false>,
                            hipFuncAttributeMaxDynamicSharedMemorySize, (int)lds);

  row_norm_kernel<<<B / 8, 256, 0, stream>>>(cnn, cn, D);
  row_norm_kernel<<<B / 8, 256, 0, stream>>>(rnn, rn, D);

  dim3 g(B / 64), blk(128);
  // run 0: scores0 = cnn . rnn^T
  fused_kernel<1, true ><<<g, blk, lds, stream>>>(cnn, rnn, cn, rn, idx, probs,
                                                  lse0, nullptr, np, nn, nullptr, B, D);
  // run 1: scores1 = rnn . cnn^T (same idx row masks)
  fused_kernel<1, false><<<g, blk, lds, stream>>>(rnn, cnn, rn, cn, idx, probs,
                                                  lse1, nullptr, np, nn, nullptr, B, D);
  fused_kernel<2, false><<<g, blk, lds, stream>>>(cnn, rnn, cn, rn, idx, probs,
                                                  nullptr, lse0, np, nn, rl0, B, D);
  fused_kernel<2, false><<<g, blk, lds, stream>>>(rnn, cnn, rn, cn, idx, probs,
                                                  nullptr, lse1, np, nn, rl1, B, D);

  finalize_kernel<<<1, 256, 0, stream>>>(rl0, rl1, (float*)d_out, B);
}